// SoftAvULoss_76209899700732
// MI455X (gfx1250) — compile-verified
//
#include <hip/hip_runtime.h>
#include <hip/hip_bf16.h>
#include <math.h>

// SoftAvU loss: N=524288 rows, C=100 classes, fp32 logits, int labels.
// HBM-bound streaming reduction: one pass over 210 MB, ~9 us floor at 23.3 TB/s.
// Hot path works in base-2: d2 = (x-m)*log2e via one FMA, e2 = v_exp_f32(d2),
// entropy = ln2 * (log2(s) - dot2/s).

namespace {

constexpr int kC   = 100;  // classes per row
constexpr int kCV4 = 25;   // 100 floats = 25 float4 (rows are 400B -> 16B aligned)

constexpr float kLOG2E = 1.4426950408889634f;  // log2(e)
constexpr float kLN2   = 0.6931471805599453f;  // ln(2)

// Native gfx1250 transcendentals: v_exp_f32 computes 2^x, v_log_f32 computes log2(x).
__device__ __forceinline__ float fast_exp2(float x) { return __builtin_amdgcn_exp2f(x); }
__device__ __forceinline__ float fast_log2(float x) { return __builtin_amdgcn_logf(x); }

__global__ __launch_bounds__(64) void avu_init_kernel(double* acc) {
  if (threadIdx.x < 2) acc[threadIdx.x] = 0.0;
}

__global__ __launch_bounds__(256) void avu_main_kernel(
    const float* __restrict__ logits, const int* __restrict__ labels,
    double* __restrict__ acc, int N) {
  __shared__ double s_acc[2];
  if (threadIdx.x == 0) { s_acc[0] = 0.0; s_acc[1] = 0.0; }
  __syncthreads();

  double num = 0.0, den = 0.0;
  const float4* lp = reinterpret_cast<const float4*>(logits);
  const long stride = (long)gridDim.x * blockDim.x;

  for (long row = (long)blockIdx.x * blockDim.x + threadIdx.x; row < N; row += stride) {
    const long base4 = row * kCV4;

    // Prefetch next row this thread will process (global_prefetch_b8 path).
    if (row + stride < (long)N) {
      __builtin_prefetch(reinterpret_cast<const void*>(lp + base4 + stride * kCV4), 0, 0);
    }

    // Load full row into registers: 25 x global_load_b128.
    float4 x[kCV4];
#pragma unroll
    for (int j = 0; j < kCV4; ++j) x[j] = lp[base4 + j];

    // Pass A: row max (v_max_f32 / v_max3_f32 chains).
    float m = x[0].x;
#pragma unroll
    for (int j = 0; j < kCV4; ++j) {
      m = fmaxf(m, fmaxf(fmaxf(x[j].x, x[j].y), fmaxf(x[j].z, x[j].w)));
    }

    // Pass B (base-2): d2 = fma(x, log2e, -m*log2e); e2 = 2^d2 (v_exp_f32);
    //   s = sum(e2) = sum(exp(x-m));  dot2 = sum(e2*d2) = log2e * sum(e*(x-m)).
    const float mneg = -m * kLOG2E;
    float s = 0.0f, dot2 = 0.0f;
#pragma unroll
    for (int j = 0; j < kCV4; ++j) {
      const float d0 = __fmaf_rn(x[j].x, kLOG2E, mneg);
      const float d1 = __fmaf_rn(x[j].y, kLOG2E, mneg);
      const float d2 = __fmaf_rn(x[j].z, kLOG2E, mneg);
      const float d3 = __fmaf_rn(x[j].w, kLOG2E, mneg);
      const float e0 = fast_exp2(d0), e1 = fast_exp2(d1);
      const float e2 = fast_exp2(d2), e3 = fast_exp2(d3);
      s += (e0 + e1) + (e2 + e3);
      dot2 = __fmaf_rn(e0, d0, dot2);
      dot2 = __fmaf_rn(e1, d1, dot2);
      dot2 = __fmaf_rn(e2, d2, dot2);
      dot2 = __fmaf_rn(e3, d3, dot2);
    }

    // entropy = -sum p*log(p) = ln2 * (log2(s) - dot2/s)  (EPS term ~1e-8, below fp32 noise)
    const float ent = kLN2 * (fast_log2(s) - dot2 / s);

    // accurate <=> logits[row, label] equals the row max (max is bit-exact an element)
    const int   lab  = labels[row];
    const float xl   = logits[row * (long)kC + lab];
    const float accf = (xl == m) ? 1.0f : 0.0f;

    // soft uncertainty: sigmoid((1/TEMP) * log(en*(1-K)/((1-en)*K))), TEMP=0.01, K=0.5
    const float en = ent * (1.0f / (float)kC);
    const float z  = 100.0f * kLN2 * fast_log2((en * 0.5f) / ((1.0f - en) * 0.5f));
    const float su = 1.0f / (1.0f + fast_exp2(-z * kLOG2E));
    const float t  = tanhf(ent);

    const float w_c = (1.0f - su) * (1.0f - t);  // certain & confident weight
    const float w_u = su * t;                    // uncertain & high-entropy weight
    num += (double)(accf * w_c + (1.0f - accf) * w_u);  // n_ac + n_iu
    den += (double)(w_c + w_u);                         // n_ac + n_au + n_ic + n_iu
  }

  // Block reduction: ds_add_f64 into LDS, then one global_atomic_add_f64 pair per block.
  atomicAdd(&s_acc[0], num);
  atomicAdd(&s_acc[1], den);
  __syncthreads();
  if (threadIdx.x == 0) {
    atomicAdd(&acc[0], s_acc[0]);
    atomicAdd(&acc[1], s_acc[1]);
  }
}

__global__ void avu_finalize_kernel(const double* __restrict__ acc,
                                    float* __restrict__ out) {
  const double avu = acc[0] / (acc[1] + 1e-10);
  out[0] = (float)(-log(avu + 1e-10));
}

}  // namespace

extern "C" void kernel_launch(void* const* d_in, const int* in_sizes, int n_in,
                              void* d_out, int out_size, void* d_ws, size_t ws_size,
                              hipStream_t stream) {
  const float* logits = (const float*)d_in[0];
  const int*   labels = (const int*)d_in[1];
  float*  out = (float*)d_out;
  double* acc = (double*)d_ws;  // acc[0] = num, acc[1] = den

  const int N = in_sizes[1];  // labels count = number of rows

  // Zero the fp64 accumulators every call (workspace is poisoned once, never re-zeroed).
  avu_init_kernel<<<1, 64, 0, stream>>>(acc);

  // 2 rows per thread so the next-row prefetch overlaps compute with HBM fetch.
  const int threads = 256;
  int blocks = (N + threads * 2 - 1) / (threads * 2);
  if (blocks < 1) blocks = 1;
  avu_main_kernel<<<blocks, threads, 0, stream>>>(logits, labels, acc, N);

  avu_finalize_kernel<<<1, 1, 0, stream>>>(acc, out);
}